// PoseSenceFlowModule_1726576853121
// MI455X (gfx1250) — compile-verified
//
#include <hip/hip_runtime.h>
#include <math.h>

// ---------------------------------------------------------------------------
// Types for f32 WMMA (V_WMMA_F32_16X16X4_F32): A=16x4 (2 VGPR), B=4x16 (2 VGPR),
// C/D=16x16 f32 (8 VGPR). Wave32 only.
// ---------------------------------------------------------------------------
typedef float v2f __attribute__((ext_vector_type(2)));
typedef float v8f __attribute__((ext_vector_type(8)));

static inline int nblk(int total, int tpb = 256) { return (total + tpb - 1) / tpb; }

// ---------------------------------------------------------------------------
// Strided subsample: out[b,m,:] = in[b, m*stride, :]
// ---------------------------------------------------------------------------
__global__ void subsample_kernel(const float* __restrict__ in, float* __restrict__ out,
                                 int N, int M, int stride, int total) {
  int t = blockIdx.x * blockDim.x + threadIdx.x;
  if (t >= total) return;                       // total = B*M*3
  int c = t % 3;
  int m = (t / 3) % M;
  int b = t / (3 * M);
  out[t] = in[((size_t)b * N + (size_t)m * stride) * 3 + c];
}

// ---------------------------------------------------------------------------
// Brute-force KNN: per query thread keeps sorted top-K (ascending sq-dist) in
// registers via unrolled insertion. Uses the same |q|^2+|r|^2-2q.r expansion
// as the reference. Ties keep the earlier index (matches lax.top_k stability).
// ---------------------------------------------------------------------------
template <int K>
__global__ void knn_kernel(const float* __restrict__ q, const float* __restrict__ ref,
                           int* __restrict__ idx, int M, int N, int total) {
  int t = blockIdx.x * blockDim.x + threadIdx.x;
  if (t >= total) return;                       // total = B*M
  int b = t / M;
  const float* qp = q + (size_t)t * 3;
  float qx = qp[0], qy = qp[1], qz = qp[2];
  float qq = qx * qx + qy * qy + qz * qz;
  const float* rp = ref + (size_t)b * N * 3;
  float best[K];
  int bi[K];
#pragma unroll
  for (int k = 0; k < K; ++k) { best[k] = 3.4e38f; bi[k] = 0; }
  for (int j = 0; j < N; ++j) {
    float rx = rp[j * 3 + 0], ry = rp[j * 3 + 1], rz = rp[j * 3 + 2];
    float d = qq + (rx * rx + ry * ry + rz * rz) - 2.0f * (qx * rx + qy * ry + qz * rz);
    if (d < best[K - 1]) {
      int p = K - 1;
#pragma unroll
      for (int s = K - 1; s >= 1; --s) {
        if (d < best[s - 1]) { best[s] = best[s - 1]; bi[s] = bi[s - 1]; p = s - 1; }
      }
      best[p] = d;
      bi[p] = j;
    }
  }
#pragma unroll
  for (int k = 0; k < K; ++k) idx[(size_t)t * K + k] = bi[k];
}

static void launch_knn(const float* q, const float* ref, int* idx,
                       int B, int M, int N, int K, hipStream_t s) {
  int total = B * M;
  int g = nblk(total, 128);
  switch (K) {
    case 4:  knn_kernel<4><<<g, 128, 0, s>>>(q, ref, idx, M, N, total); break;
    case 6:  knn_kernel<6><<<g, 128, 0, s>>>(q, ref, idx, M, N, total); break;
    case 8:  knn_kernel<8><<<g, 128, 0, s>>>(q, ref, idx, M, N, total); break;
    case 16: knn_kernel<16><<<g, 128, 0, s>>>(q, ref, idx, M, N, total); break;
    default: knn_kernel<32><<<g, 128, 0, s>>>(q, ref, idx, M, N, total); break;
  }
}

// ---------------------------------------------------------------------------
// Grouping: out[row] = xyz_first ? [r.xyz[j]-q.xyz, r.feat[j]] : [r.feat[j], d]
// row = b*M*S + m*S + s ; optionally also dumps d into dout for attention.
// ---------------------------------------------------------------------------
__global__ void group_kernel(const float* __restrict__ qxyz, const float* __restrict__ rxyz,
                             const float* __restrict__ rfeat, const int* __restrict__ idx,
                             float* __restrict__ out, float* __restrict__ dout,
                             int M, int S, int Nr, int C, int xyz_first, int total) {
  int t = blockIdx.x * blockDim.x + threadIdx.x;
  if (t >= total) return;                       // total = B*M*S
  int bm = t / S;
  int b = bm / M;
  int j = idx[t];
  const float* qp = qxyz + (size_t)bm * 3;
  const float* rp = rxyz + ((size_t)b * Nr + j) * 3;
  float d0 = rp[0] - qp[0], d1 = rp[1] - qp[1], d2 = rp[2] - qp[2];
  int D = C + 3;
  float* o = out + (size_t)t * D;
  const float* f = rfeat + ((size_t)b * Nr + j) * C;
  if (xyz_first) {
    o[0] = d0; o[1] = d1; o[2] = d2;
    for (int c = 0; c < C; ++c) o[3 + c] = f[c];
  } else {
    for (int c = 0; c < C; ++c) o[c] = f[c];
    o[C] = d0; o[C + 1] = d1; o[C + 2] = d2;
  }
  if (dout) { dout[(size_t)t * 3 + 0] = d0; dout[(size_t)t * 3 + 1] = d1; dout[(size_t)t * 3 + 2] = d2; }
}

// Cost-volume stage-1 grouping: [gather(feat2), broadcast(feat1), d]
__global__ void cv_group1_kernel(const float* __restrict__ xyz1, const float* __restrict__ f1,
                                 const float* __restrict__ xyz2, const float* __restrict__ f2,
                                 const int* __restrict__ idx, float* __restrict__ out,
                                 float* __restrict__ dout, int M, int S, int N2, int C, int total) {
  int t = blockIdx.x * blockDim.x + threadIdx.x;
  if (t >= total) return;                       // total = B*M*S
  int bm = t / S;
  int b = bm / M;
  int j = idx[t];
  const float* qp = xyz1 + (size_t)bm * 3;
  const float* rp = xyz2 + ((size_t)b * N2 + j) * 3;
  float d0 = rp[0] - qp[0], d1 = rp[1] - qp[1], d2 = rp[2] - qp[2];
  int D = 2 * C + 3;
  float* o = out + (size_t)t * D;
  const float* fa = f2 + ((size_t)b * N2 + j) * C;
  const float* fb = f1 + (size_t)bm * C;
  for (int c = 0; c < C; ++c) o[c] = fa[c];
  for (int c = 0; c < C; ++c) o[C + c] = fb[c];
  o[2 * C] = d0; o[2 * C + 1] = d1; o[2 * C + 2] = d2;
  dout[(size_t)t * 3 + 0] = d0; dout[(size_t)t * 3 + 1] = d1; dout[(size_t)t * 3 + 2] = d2;
}

// ---------------------------------------------------------------------------
// f32 WMMA GEMM: Y[R,N] = act(X[R,K] @ W[K,N] + bias).
// One wave = one 16x16 output tile; K consumed 4/step via V_WMMA_F32_16X16X4_F32.
//
// Out-of-range rows/cols need NO masking: each D[m,n] is an independent dot
// product, so garbage A-rows / B-cols only affect D entries we never store.
// We therefore CLAMP addresses (always-in-bounds unconditional loads, no EXEC
// churn) and only mask the K-overflow tail lanes (they would pollute valid
// outputs). Stores take an unguarded fast path for interior tiles.
// ---------------------------------------------------------------------------
__global__ void gemm_kernel(const float* __restrict__ X, const float* __restrict__ W,
                            const float* __restrict__ bias, float* __restrict__ Y,
                            int R, int K, int N, int relu, int rowTiles) {
  int rt = blockIdx.x * 4 + threadIdx.y;
  if (rt >= rowTiles) return;                   // wave-uniform
  int m0 = rt * 16;
  int n0 = blockIdx.y * 16;
  int lane = threadIdx.x;                       // 0..31 (wave32)
  int lm = lane & 15;
  int lh = lane >> 4;
  int row = m0 + lm;
  int col = n0 + lm;
  int rowc = (row < R) ? row : (R - 1);         // clamped: loads always legal
  int colc = (col < N) ? col : (N - 1);
  const float* xp = X + (size_t)rowc * K + lh * 2;   // A frag ptr (k = lh*2 + {0,1})
  const float* wp = W + (size_t)(lh * 2) * N + colc; // B frag ptr
  v8f acc = {};
  int Kmain = K & ~3;
  for (int k0 = 0; k0 < Kmain; k0 += 4) {
    v2f a, b;
    a.x = xp[0];
    a.y = xp[1];
    b.x = wp[0];
    b.y = wp[(size_t)N];
    xp += 4;
    wp += (size_t)4 * N;
    acc = __builtin_amdgcn_wmma_f32_16x16x4_f32(false, a, false, b, (short)0, acc, false, false);
  }
  if (K & 3) {                                  // uniform tail: mask k-overflow lanes
    int ka = Kmain + lh * 2;
    int k0c = (ka < K) ? ka : (K - 1);
    int k1c = (ka + 1 < K) ? (ka + 1) : (K - 1);
    const float* Xr = X + (size_t)rowc * K;
    float ax = Xr[k0c];
    float ay = Xr[k1c];
    float bx = W[(size_t)k0c * N + colc];
    float by = W[(size_t)k1c * N + colc];
    v2f a, b;
    a.x = (ka < K) ? ax : 0.0f;                 // cndmask, not branch
    a.y = (ka + 1 < K) ? ay : 0.0f;
    b.x = (ka < K) ? bx : 0.0f;
    b.y = (ka + 1 < K) ? by : 0.0f;
    acc = __builtin_amdgcn_wmma_f32_16x16x4_f32(false, a, false, b, (short)0, acc, false, false);
  }
  float bv = bias[colc];
  if (m0 + 16 <= R && n0 + 16 <= N) {           // interior tile: unguarded stores
    float* yp = Y + (size_t)(m0 + lh * 8) * N + col;
#pragma unroll
    for (int r = 0; r < 8; ++r) {
      float v = acc[r] + bv;
      if (relu) v = fmaxf(v, 0.0f);
      yp[(size_t)r * N] = v;
    }
  } else {                                      // edge tile: guarded stores
    bool cok = col < N;
#pragma unroll
    for (int r = 0; r < 8; ++r) {
      int orow = m0 + r + lh * 8;               // D layout: vgpr r -> M=r / r+8
      if ((orow < R) && cok) {
        float v = acc[r] + bv;
        if (relu) v = fmaxf(v, 0.0f);
        Y[(size_t)orow * N + col] = v;
      }
    }
  }
}

static void launch_gemm(const float* X, const float* W, const float* bias, float* Y,
                        int R, int K, int N, int relu, hipStream_t s) {
  int rowTiles = (R + 15) / 16;
  int colTiles = (N + 15) / 16;
  dim3 grid((rowTiles + 3) / 4, colTiles);
  dim3 block(32, 4);
  gemm_kernel<<<grid, block, 0, s>>>(X, W, bias, Y, R, K, N, relu, rowTiles);
}

// ---------------------------------------------------------------------------
// Max-pool over the neighbor axis: in[BM*S, C] -> out[BM, C]
// ---------------------------------------------------------------------------
__global__ void maxpool_kernel(const float* __restrict__ in, float* __restrict__ out,
                               int S, int C, int total) {
  int t = blockIdx.x * blockDim.x + threadIdx.x;
  if (t >= total) return;                       // total = BM*C
  int bm = t / C;
  int c = t % C;
  const float* p = in + (size_t)bm * S * C + c;
  float v = -3.4e38f;
  for (int s = 0; s < S; ++s) v = fmaxf(v, p[(size_t)s * C]);
  out[t] = v;
}

// ---------------------------------------------------------------------------
// softmax over S (per (bm,c) channel) applied as weights on h, then sum:
// out[bm,c] = sum_s softmax_s(logits[bm,s,c]) * h[bm,s,c]
// ---------------------------------------------------------------------------
__global__ void softmax_wsum_kernel(const float* __restrict__ logits, const float* __restrict__ h,
                                    float* __restrict__ out, int S, int C, int total) {
  int t = blockIdx.x * blockDim.x + threadIdx.x;
  if (t >= total) return;                       // total = BM*C
  int bm = t / C;
  int c = t % C;
  const float* lp = logits + (size_t)bm * S * C + c;
  const float* hp = h + (size_t)bm * S * C + c;
  float mx = -3.4e38f;
  for (int s = 0; s < S; ++s) mx = fmaxf(mx, lp[(size_t)s * C]);
  float sum = 0.0f, acc = 0.0f;
  for (int s = 0; s < S; ++s) {
    float e = expf(lp[(size_t)s * C] - mx);
    sum += e;
    acc += e * hp[(size_t)s * C];
  }
  out[t] = acc / sum;
}

__global__ void concat2_kernel(const float* __restrict__ a, const float* __restrict__ b,
                               float* __restrict__ out, int Ca, int Cb, int total) {
  int t = blockIdx.x * blockDim.x + threadIdx.x;
  if (t >= total) return;                       // total = R*(Ca+Cb)
  int D = Ca + Cb;
  int r = t / D;
  int c = t % D;
  out[t] = (c < Ca) ? a[(size_t)r * Ca + c] : b[(size_t)r * Cb + (c - Ca)];
}

// ---------------------------------------------------------------------------
// SA module: subsample -> KNN -> group -> 3xGEMM(relu) -> maxpool
// ---------------------------------------------------------------------------
static void sa_stage(const float* xyz, const float* feat, int B, int Nin, int C,
                     int npoint, int nsample,
                     const float* w0, const float* b0, int c0,
                     const float* w1, const float* b1, int c1,
                     const float* w2, const float* b2, int c2,
                     float* out_xyz, float* out_feat,
                     float* arA, float* arB, int* idxb, hipStream_t s) {
  int stride = Nin / npoint;
  int t1 = B * npoint * 3;
  subsample_kernel<<<nblk(t1), 256, 0, s>>>(xyz, out_xyz, Nin, npoint, stride, t1);
  launch_knn(out_xyz, xyz, idxb, B, npoint, Nin, nsample, s);
  int rows = B * npoint * nsample;
  group_kernel<<<nblk(rows), 256, 0, s>>>(out_xyz, xyz, feat, idxb, arA, nullptr,
                                          npoint, nsample, Nin, C, 1, rows);
  launch_gemm(arA, w0, b0, arB, rows, C + 3, c0, 1, s);
  launch_gemm(arB, w1, b1, arA, rows, c0, c1, 1, s);
  launch_gemm(arA, w2, b2, arB, rows, c1, c2, 1, s);
  int t2 = B * npoint * c2;
  maxpool_kernel<<<nblk(t2), 256, 0, s>>>(arB, out_feat, nsample, c2, t2);
}

// ---------------------------------------------------------------------------
// kernel_launch
// ---------------------------------------------------------------------------
extern "C" void kernel_launch(void* const* d_in, const int* in_sizes, int n_in,
                              void* d_out, int out_size, void* d_ws, size_t ws_size,
                              hipStream_t stream) {
  const int B = 4, N = 8192;
  const float* xyz1   = (const float*)d_in[0];
  const float* xyz2   = (const float*)d_in[1];
  const float* color1 = (const float*)d_in[2];
  const float* color2 = (const float*)d_in[3];
  // params, in setup_inputs() dict insertion order
  int p = 4;
  const float* sa_w[4][3]; const float* sa_b[4][3];
  for (int l = 0; l < 4; ++l)
    for (int i = 0; i < 3; ++i) { sa_w[l][i] = (const float*)d_in[p++]; sa_b[l][i] = (const float*)d_in[p++]; }
  const float* cv1_w[3]; const float* cv1_b[3];
  for (int i = 0; i < 3; ++i) { cv1_w[i] = (const float*)d_in[p++]; cv1_b[i] = (const float*)d_in[p++]; }
  const float* cvq1_w = (const float*)d_in[p++]; const float* cvq1_b = (const float*)d_in[p++];
  const float* cv2_w[2]; const float* cv2_b[2];
  for (int i = 0; i < 2; ++i) { cv2_w[i] = (const float*)d_in[p++]; cv2_b[i] = (const float*)d_in[p++]; }
  const float* cvq2_w = (const float*)d_in[p++]; const float* cvq2_b = (const float*)d_in[p++];
  const float* up_w[3]; const float* up_b[3];
  for (int i = 0; i < 3; ++i) { up_w[i] = (const float*)d_in[p++]; up_b[i] = (const float*)d_in[p++]; }
  const float* up2_w = (const float*)d_in[p++]; const float* up2_b = (const float*)d_in[p++];
  const float* fp_w  = (const float*)d_in[p++]; const float* fp_b  = (const float*)d_in[p++];
  const float* fl_w  = (const float*)d_in[p++]; const float* fl_b  = (const float*)d_in[p++];

  // ---- workspace carve (floats), deterministic every call ----
  float* ws = (float*)d_ws;
  size_t off = 0;
  auto take = [&](size_t n) { float* q = ws + off; off += n; return q; };
  float* arA    = take(4200000);           // max grouped tensor 262144x16
  float* arB    = take(4200000);
  float* dbuf   = take(8192);              // xyz diffs for attention (<=4608)
  float* logits = take(200704);            // attention logits (<=196608)
  int*   idxb   = (int*)take(262144);      // KNN indices (<= B*2048*32)
  float* l0x1 = take(B*2048*3);  float* l0f1 = take(B*2048*16);
  float* l0x2 = take(B*2048*3);  float* l0f2 = take(B*2048*16);
  float* l1x1 = take(B*1024*3);  float* l1f1 = take(B*1024*32);
  float* l1x2 = take(B*1024*3);  float* l1f2 = take(B*1024*32);
  float* l2x1 = take(B*256*3);   float* l2f1 = take(B*256*64);
  float* l2x2 = take(B*256*3);   float* l2f2 = take(B*256*64);
  float* l3x1 = take(B*64*3);    float* l3f1 = take(B*64*128);
  float* l3x2 = take(B*64*3);    float* l3f2 = take(B*64*128);
  float* pc    = take(B*64*128);
  float* cost3 = take(B*64*128);
  float* hup   = take(B*256*256);
  float* cat2  = take(B*256*320);
  float* up2o  = take(B*256*256);
  float* ff    = take(B*256*128);
  (void)ws_size; (void)n_in; (void)in_sizes; (void)out_size;

  // ---- SA pyramid on both clouds ----
  sa_stage(xyz1, color1, B, N,    3, 2048, 32, sa_w[0][0], sa_b[0][0], 8,  sa_w[0][1], sa_b[0][1], 8,  sa_w[0][2], sa_b[0][2], 16,  l0x1, l0f1, arA, arB, idxb, stream);
  sa_stage(xyz2, color2, B, N,    3, 2048, 32, sa_w[0][0], sa_b[0][0], 8,  sa_w[0][1], sa_b[0][1], 8,  sa_w[0][2], sa_b[0][2], 16,  l0x2, l0f2, arA, arB, idxb, stream);
  sa_stage(l0x1, l0f1,   B, 2048, 16, 1024, 32, sa_w[1][0], sa_b[1][0], 16, sa_w[1][1], sa_b[1][1], 16, sa_w[1][2], sa_b[1][2], 32,  l1x1, l1f1, arA, arB, idxb, stream);
  sa_stage(l0x2, l0f2,   B, 2048, 16, 1024, 32, sa_w[1][0], sa_b[1][0], 16, sa_w[1][1], sa_b[1][1], 16, sa_w[1][2], sa_b[1][2], 32,  l1x2, l1f2, arA, arB, idxb, stream);
  sa_stage(l1x1, l1f1,   B, 1024, 32, 256,  16, sa_w[2][0], sa_b[2][0], 32, sa_w[2][1], sa_b[2][1], 32, sa_w[2][2], sa_b[2][2], 64,  l2x1, l2f1, arA, arB, idxb, stream);
  sa_stage(l1x2, l1f2,   B, 1024, 32, 256,  16, sa_w[2][0], sa_b[2][0], 32, sa_w[2][1], sa_b[2][1], 32, sa_w[2][2], sa_b[2][2], 64,  l2x2, l2f2, arA, arB, idxb, stream);
  sa_stage(l2x1, l2f1,   B, 256,  64, 64,   16, sa_w[3][0], sa_b[3][0], 64, sa_w[3][1], sa_b[3][1], 64, sa_w[3][2], sa_b[3][2], 128, l3x1, l3f1, arA, arB, idxb, stream);
  sa_stage(l2x2, l2f2,   B, 256,  64, 64,   16, sa_w[3][0], sa_b[3][0], 64, sa_w[3][1], sa_b[3][1], 64, sa_w[3][2], sa_b[3][2], 128, l3x2, l3f2, arA, arB, idxb, stream);

  // ---- Cost volume (nq=6 across clouds, ns=4 within cloud 1) ----
  {
    int M = 64, S = 6, C = 128;
    launch_knn(l3x1, l3x2, idxb, B, M, 64, S, stream);
    int rows = B * M * S;
    cv_group1_kernel<<<nblk(rows), 256, 0, stream>>>(l3x1, l3f1, l3x2, l3f2, idxb,
                                                     arA, dbuf, M, S, 64, C, rows);
    launch_gemm(arA, cv1_w[0], cv1_b[0], arB, rows, 2 * C + 3, 256, 1, stream);
    launch_gemm(arB, cv1_w[1], cv1_b[1], arA, rows, 256, 128, 1, stream);
    launch_gemm(arA, cv1_w[2], cv1_b[2], arB, rows, 128, 128, 1, stream);   // h in arB
    launch_gemm(dbuf, cvq1_w, cvq1_b, logits, rows, 3, 128, 0, stream);
    int t2 = B * M * 128;
    softmax_wsum_kernel<<<nblk(t2), 256, 0, stream>>>(logits, arB, pc, S, 128, t2);

    S = 4;
    launch_knn(l3x1, l3x1, idxb, B, M, 64, S, stream);
    rows = B * M * S;
    group_kernel<<<nblk(rows), 256, 0, stream>>>(l3x1, l3x1, pc, idxb, arA, dbuf,
                                                 M, S, 64, 128, /*xyz_first=*/0, rows);
    launch_gemm(arA, cv2_w[0], cv2_b[0], arB, rows, 131, 256, 1, stream);
    launch_gemm(arB, cv2_w[1], cv2_b[1], arA, rows, 256, 128, 1, stream);   // h2 in arA
    launch_gemm(dbuf, cvq2_w, cvq2_b, logits, rows, 3, 128, 0, stream);
    softmax_wsum_kernel<<<nblk(t2), 256, 0, stream>>>(logits, arA, cost3, S, 128, t2);
  }

  // ---- Upconv + head ----
  {
    int M = 256, S = 8;
    launch_knn(l2x1, l3x1, idxb, B, M, 64, S, stream);
    int rows = B * M * S;
    group_kernel<<<nblk(rows), 256, 0, stream>>>(l2x1, l3x1, cost3, idxb, arA, nullptr,
                                                 M, S, 64, 128, /*xyz_first=*/1, rows);
    launch_gemm(arA, up_w[0], up_b[0], arB, rows, 131, 128, 1, stream);
    launch_gemm(arB, up_w[1], up_b[1], arA, rows, 128, 128, 1, stream);
    launch_gemm(arA, up_w[2], up_b[2], arB, rows, 128, 256, 1, stream);
    int t2 = B * M * 256;
    maxpool_kernel<<<nblk(t2), 256, 0, stream>>>(arB, hup, S, 256, t2);

    int Rr = B * M;
    int tc = Rr * 320;
    concat2_kernel<<<nblk(tc), 256, 0, stream>>>(hup, l2f1, cat2, 256, 64, tc);
    launch_gemm(cat2, up2_w, up2_b, up2o, Rr, 320, 256, 1, stream);
    launch_gemm(up2o, fp_w, fp_b, ff, Rr, 256, 128, 1, stream);
    launch_gemm(ff, fl_w, fl_b, (float*)d_out, Rr, 128, 3, 0, stream);
  }
}